// SpikesEncoder_23287312679626
// MI455X (gfx1250) — compile-verified
//
#include <hip/hip_runtime.h>

typedef __attribute__((ext_vector_type(2))) float v2f;
typedef __attribute__((ext_vector_type(8))) float v8f;

#define NN 2048                     // n_units (== K)
#define BB 256                      // batch
#define TT 128                      // n_steps
#define ALPHA_F 0.90483741803595957f  // exp(-DT/TAU_M) = exp(-0.1)

// Fused GEMM (I = x @ W via V_WMMA_F32_16X16X4_F32) + 128-step LIF scan.
// One wave owns a 16x16 tile of I; 4 waves per block cover 64 columns.
__global__ __launch_bounds__(128) void lif_fused_kernel(
    const float* __restrict__ x,   // [BB, NN]
    const float* __restrict__ W,   // [NN, NN]
    float* __restrict__ out)       // [BB, TT, NN]
{
  const int wid  = threadIdx.x >> 5;   // wave in block (0..3)
  const int lane = threadIdx.x & 31;
  const int lo   = lane & 15;          // 0..15
  const int hi   = lane >> 4;          // 0 or 1

  const int m0 = blockIdx.y * 16;                  // batch-row tile
  const int n0 = (blockIdx.x * 4 + wid) * 16;      // neuron-col tile

  // A-matrix 16x4 f32 layout: lanes 0-15 hold M=lane, K=k..k+1; lanes 16-31 K=k+2..k+3
  const float* Aptr = x + (size_t)(m0 + lo) * NN + 2 * hi;
  // B-matrix 4x16 f32 layout: lanes 0-15 hold N=lane, K=k..k+1 (vgpr0/1); lanes 16-31 K=k+2..k+3
  const float* Bptr = W + (size_t)(2 * hi) * NN + (n0 + lo);

  v8f acc = {0.f, 0.f, 0.f, 0.f, 0.f, 0.f, 0.f, 0.f};

#pragma unroll 4
  for (int k = 0; k < NN; k += 4) {
    v2f a = *(const v2f*)(Aptr + k);                 // global_load_b64
    v2f b;
    b.x = Bptr[(size_t)k * NN];                      // B[k+2hi][n]
    b.y = Bptr[(size_t)(k + 1) * NN];                // B[k+2hi+1][n]
    acc = __builtin_amdgcn_wmma_f32_16x16x4_f32(
        /*neg_a=*/false, a, /*neg_b=*/false, b,
        /*c_mod=*/(short)0, acc, /*reuse_a=*/false, /*reuse_b=*/false);
  }

  // C/D layout: acc[r] is element (M = m0 + 8*hi + r, N = n0 + lo).
  float I[8], V[8], Z[8];
#pragma unroll
  for (int r = 0; r < 8; ++r) { I[r] = acc[r]; V[r] = 0.f; Z[r] = 0.f; }

  float* outp = out + (size_t)(m0 + 8 * hi) * TT * NN + (n0 + lo);
  const size_t rstride = (size_t)TT * NN;   // advance one batch row

  for (int t = 0; t < TT; ++t) {
    float* op = outp + (size_t)t * NN;
#pragma unroll
    for (int r = 0; r < 8; ++r) {
      float v = fmaf(ALPHA_F, V[r], I[r]) * (1.0f - Z[r]);  // reset by prev spike
      float z = (v >= 1.0f) ? 1.0f : 0.0f;                  // Heaviside spike
      V[r] = v;
      Z[r] = z;
      // 256 MB streaming output > L2: non-temporal so W/x stay cached.
      __builtin_nontemporal_store(z, op + (size_t)r * rstride);
    }
  }
}

extern "C" void kernel_launch(void* const* d_in, const int* in_sizes, int n_in,
                              void* d_out, int out_size, void* d_ws, size_t ws_size,
                              hipStream_t stream) {
  const float* x = (const float*)d_in[0];   // [256, 2048] f32
  const float* W = (const float*)d_in[1];   // [2048, 2048] f32
  float* out = (float*)d_out;               // [256, 128, 2048] f32

  dim3 grid(NN / (16 * 4), BB / 16);        // 32 x 16 blocks
  dim3 block(128);                          // 4 wave32 waves
  lif_fused_kernel<<<grid, block, 0, stream>>>(x, W, out);
}